// WeightNet_13864154431858
// MI455X (gfx1250) — compile-verified
//
#include <hip/hip_runtime.h>
#include <stdint.h>
#include <math.h>

// ---------------------------------------------------------------------------
// Rowwise scaled cosine similarity on MI455X (gfx1250).
// sim[i] = scale * dot(x_i, p_i) / (max(||x_i||,eps) * max(||p_i||,eps))
// n = 131072 rows, d = 512 f32 -> pure HBM-bound streaming reduction.
// Strategy: CDNA5 async global->LDS double-buffered pipeline (ASYNCcnt),
// ds_load_b128 readback, VALU f32 FMAs, wave32 shuffle reduction.
// ---------------------------------------------------------------------------

#define DIM          512
#define ROW_BYTES    (DIM * 4)            // 2048 B per row per matrix
#define THREADS      256
#define WAVES_PER_BLOCK (THREADS / 32)    // 8 waves (wave32!)
#define F4_PER_ROW   (DIM / 4)            // 128 float4 per row
#define BUF_F4       (2 * F4_PER_ROW)     // x-row + p-row = 256 float4 = 4 KB
#define WAVE_F4      (2 * BUF_F4)         // double buffer = 512 float4 = 8 KB/wave
#define BLOCKS       2048

static __device__ __forceinline__ uint32_t lds_offset(const void* q) {
    // Generic pointer to LDS: low 32 bits are the wave-relative LDS byte address
    // (aperture form {SHARED_BASE[63:32], lds_off[31:0]}), which is exactly what
    // the async-to-LDS VDST VGPR expects.
    return (uint32_t)(uintptr_t)q;
}

// Issue one row (x + p) into an LDS buffer: 8 async b128 copies, 512 B each.
// INST_OFFSET applies to BOTH the LDS and global addresses (ISA 08, §4.4),
// so one base address per lane covers all 4 chunks of a 2 KB row.
static __device__ __forceinline__ void issue_row(const float* __restrict__ x,
                                                 const float* __restrict__ p,
                                                 int row,
                                                 uint32_t lds_x, uint32_t lds_p,
                                                 int lane) {
    uint64_t gx = (uint64_t)(uintptr_t)x + (uint64_t)row * ROW_BYTES + (uint32_t)(lane * 16);
    uint64_t gp = (uint64_t)(uintptr_t)p + (uint64_t)row * ROW_BYTES + (uint32_t)(lane * 16);
    uint32_t lx = lds_x + (uint32_t)(lane * 16);
    uint32_t lp = lds_p + (uint32_t)(lane * 16);
    asm volatile("global_load_async_to_lds_b128 %0, %1, off"             :: "v"(lx), "v"(gx) : "memory");
    asm volatile("global_load_async_to_lds_b128 %0, %1, off offset:512"  :: "v"(lx), "v"(gx) : "memory");
    asm volatile("global_load_async_to_lds_b128 %0, %1, off offset:1024" :: "v"(lx), "v"(gx) : "memory");
    asm volatile("global_load_async_to_lds_b128 %0, %1, off offset:1536" :: "v"(lx), "v"(gx) : "memory");
    asm volatile("global_load_async_to_lds_b128 %0, %1, off"             :: "v"(lp), "v"(gp) : "memory");
    asm volatile("global_load_async_to_lds_b128 %0, %1, off offset:512"  :: "v"(lp), "v"(gp) : "memory");
    asm volatile("global_load_async_to_lds_b128 %0, %1, off offset:1024" :: "v"(lp), "v"(gp) : "memory");
    asm volatile("global_load_async_to_lds_b128 %0, %1, off offset:1536" :: "v"(lp), "v"(gp) : "memory");
}

static __device__ __forceinline__ void wait_async_le8() {
    asm volatile("s_wait_asynccnt 8" ::: "memory");   // one 8-op batch may stay in flight
}
static __device__ __forceinline__ void wait_async_0() {
    asm volatile("s_wait_asynccnt 0" ::: "memory");
}
static __device__ __forceinline__ void wait_ds_0() {
    // Async-to-LDS writes are unordered vs DS reads: drain our ds_loads before
    // letting the DMA overwrite a buffer we just consumed.
    asm volatile("s_wait_dscnt 0" ::: "memory");
}

__global__ __launch_bounds__(THREADS)
void cosine_sim_kernel(const float* __restrict__ x,
                       const float* __restrict__ p,
                       const float* __restrict__ scale_ptr,
                       float* __restrict__ out,
                       int nrows) {
    __shared__ float4 smem[WAVES_PER_BLOCK * WAVE_F4];  // 64 KB / block

    const int lane   = (int)(threadIdx.x & 31);
    const int wave   = (int)(threadIdx.x >> 5);
    const int gwave  = (int)(blockIdx.x) * WAVES_PER_BLOCK + wave;
    const int nwaves = (int)(gridDim.x) * WAVES_PER_BLOCK;

    float4* wbuf = &smem[wave * WAVE_F4];
    const uint32_t lds_base = lds_offset(wbuf);
    const float scale = *scale_ptr;

    // Prologue: fill both buffers (rows gwave, gwave + nwaves).
    if (gwave < nrows)
        issue_row(x, p, gwave, lds_base, lds_base + F4_PER_ROW * 16, lane);
    if (gwave + nwaves < nrows)
        issue_row(x, p, gwave + nwaves,
                  lds_base + BUF_F4 * 16,
                  lds_base + BUF_F4 * 16 + F4_PER_ROW * 16, lane);

    int k = 0;
    for (int row = gwave; row < nrows; row += nwaves, ++k) {
        const bool more = (row + nwaves) < nrows;
        if (more) wait_async_le8(); else wait_async_0();   // batch for `row` is done

        const float4* bx = wbuf + (k & 1) * BUF_F4;
        const float4* bp = bx + F4_PER_ROW;

        float xp = 0.0f, xx = 0.0f, pp = 0.0f;
        #pragma unroll
        for (int j = 0; j < 4; ++j) {
            float4 a = bx[j * 32 + lane];   // ds_load_b128
            float4 b = bp[j * 32 + lane];   // ds_load_b128
            xp = fmaf(a.x, b.x, xp); xp = fmaf(a.y, b.y, xp);
            xp = fmaf(a.z, b.z, xp); xp = fmaf(a.w, b.w, xp);
            xx = fmaf(a.x, a.x, xx); xx = fmaf(a.y, a.y, xx);
            xx = fmaf(a.z, a.z, xx); xx = fmaf(a.w, a.w, xx);
            pp = fmaf(b.x, b.x, pp); pp = fmaf(b.y, b.y, pp);
            pp = fmaf(b.z, b.z, pp); pp = fmaf(b.w, b.w, pp);
        }

        // Refill this buffer with the row two steps ahead.
        long next2 = (long)row + 2L * (long)nwaves;
        if (next2 < (long)nrows) {
            wait_ds_0();
            uint32_t lb = lds_base + (uint32_t)((k & 1) * (BUF_F4 * 16));
            issue_row(x, p, (int)next2, lb, lb + F4_PER_ROW * 16, lane);
        }

        // Wave32 cross-lane reduction (5 steps, explicit width 32).
        #pragma unroll
        for (int off = 16; off > 0; off >>= 1) {
            xp += __shfl_xor(xp, off, 32);
            xx += __shfl_xor(xx, off, 32);
            pp += __shfl_xor(pp, off, 32);
        }

        if (lane == 0) {
            float nx = fmaxf(sqrtf(xx), 1e-12f);
            float np = fmaxf(sqrtf(pp), 1e-12f);
            out[row] = scale * xp / (nx * np);
        }
    }
}

extern "C" void kernel_launch(void* const* d_in, const int* in_sizes, int n_in,
                              void* d_out, int out_size, void* d_ws, size_t ws_size,
                              hipStream_t stream) {
    const float* x     = (const float*)d_in[0];
    const float* p     = (const float*)d_in[1];
    const float* scale = (const float*)d_in[2];
    float* out = (float*)d_out;

    const int nrows = in_sizes[0] / DIM;   // 131072 for the reference shapes

    // 2048 blocks x 256 threads = 16384 waves -> 8 rows/wave for n=131072.
    // 64 KB LDS/block -> up to 5 resident blocks per 320 KB WGP.
    hipLaunchKernelGGL(cosine_sim_kernel, dim3(BLOCKS), dim3(THREADS), 0, stream,
                       x, p, scale, out, nrows);
}